// ChebNet_41807211659450
// MI455X (gfx1250) — compile-verified
//
#include <hip/hip_runtime.h>
#include <stdint.h>

typedef float v2f __attribute__((ext_vector_type(2)));
typedef float v8f __attribute__((ext_vector_type(8)));
typedef unsigned int u32x4 __attribute__((ext_vector_type(4)));
typedef int i32x4 __attribute__((ext_vector_type(4)));
typedef int i32x8 __attribute__((ext_vector_type(8)));

#define NNODES 50000

// ---------------------------------------------------------------- utilities

__global__ void cheb_fill_zero(float* __restrict__ p, long n) {
    long i = (long)blockIdx.x * blockDim.x + threadIdx.x;
    long stride = (long)gridDim.x * blockDim.x;
    for (; i < n; i += stride) p[i] = 0.0f;
}

__global__ void cheb_degree(const int* __restrict__ dst, float* __restrict__ deg, int E) {
    int e = blockIdx.x * blockDim.x + threadIdx.x;
    if (e < E) unsafeAtomicAdd(&deg[dst[e]], 1.0f);
}

__global__ void cheb_norm(float* __restrict__ nrm, int N) {
    int i = blockIdx.x * blockDim.x + threadIdx.x;
    if (i < N) {
        float d = nrm[i];
        d = d < 1.0f ? 1.0f : d;
        nrm[i] = rsqrtf(d);
    }
}

// copy X (row stride ldx) into Y (row stride ldy), float4 granularity
__global__ void cheb_copy_block(const float* __restrict__ x, int ldx,
                                float* __restrict__ y, int ldy,
                                long total, int lg) {
    long idx = (long)blockIdx.x * blockDim.x + threadIdx.x;
    if (idx >= total) return;
    long i = idx >> lg;
    int g = (int)(idx & ((1 << lg) - 1)) * 4;
    *(float4*)(y + (size_t)i * ldy + g) = *(const float4*)(x + (size_t)i * ldx + g);
}

// agg[dst[e], :] += x[src[e], :] * norm[src[e]]   (float4 per thread)
__global__ void cheb_scatter(const float* __restrict__ x, int ldx,
                             const int* __restrict__ src, const int* __restrict__ dst,
                             const float* __restrict__ nrm,
                             float* __restrict__ agg, int lda,
                             long total, int lg) {
    long idx = (long)blockIdx.x * blockDim.x + threadIdx.x;
    if (idx >= total) return;
    long e = idx >> lg;
    int g = (int)(idx & ((1 << lg) - 1)) * 4;
    int s = src[e];
    int d = dst[e];
    float ns = nrm[s];
    float4 v = *(const float4*)(x + (size_t)s * ldx + g);
    float* p = agg + (size_t)d * lda + g;
    unsafeAtomicAdd(p + 0, v.x * ns);
    unsafeAtomicAdd(p + 1, v.y * ns);
    unsafeAtomicAdd(p + 2, v.z * ns);
    unsafeAtomicAdd(p + 3, v.w * ns);
}

// T1 = -norm[i] * agg   (in place)
__global__ void cheb_finalize_t1(float* __restrict__ t, int ld,
                                 const float* __restrict__ nrm,
                                 long total, int lg) {
    long idx = (long)blockIdx.x * blockDim.x + threadIdx.x;
    if (idx >= total) return;
    long i = idx >> lg;
    int g = (int)(idx & ((1 << lg) - 1)) * 4;
    float s = -nrm[i];
    float4* p = (float4*)(t + (size_t)i * ld + g);
    float4 v = *p;
    v.x *= s; v.y *= s; v.z *= s; v.w *= s;
    *p = v;
}

// T2 = -2*norm[i]*agg - T0   (in place on agg block)
__global__ void cheb_finalize_t2(float* __restrict__ t2, const float* __restrict__ t0,
                                 int ld, const float* __restrict__ nrm,
                                 long total, int lg) {
    long idx = (long)blockIdx.x * blockDim.x + threadIdx.x;
    if (idx >= total) return;
    long i = idx >> lg;
    int g = (int)(idx & ((1 << lg) - 1)) * 4;
    float s = -2.0f * nrm[i];
    float4* p2 = (float4*)(t2 + (size_t)i * ld + g);
    const float4 v0 = *(const float4*)(t0 + (size_t)i * ld + g);
    float4 v = *p2;
    v.x = s * v.x - v0.x;
    v.y = s * v.y - v0.y;
    v.z = s * v.z - v0.z;
    v.w = s * v.w - v0.w;
    *p2 = v;
}

// ---------------------------------------------------------------- WMMA GEMM
// H[N, Fo] = A[N, K] (row stride lda) @ W[K, Fo] + bias
// Block: 16 rows (M tile). Wave w: 16-column N tile n0 = 16*w.
// A chunk (16 x 64 f32) staged into LDS by the Tensor Data Mover; TDM pad
// feature inserts 4 DWORDs after every 64 DWORDs -> LDS row stride 68 floats
// (bank-conflict-free for the 16-lane fragment reads). f32 WMMA 16x16x4.
__global__ void cheb_gemm_wmma(const float* __restrict__ A, int lda, int K,
                               const float* __restrict__ W,
                               const float* __restrict__ bias,
                               float* __restrict__ H, int Fo) {
    __shared__ float lds[16 * 68];
    const int lane = threadIdx.x & 31;
    const int lr   = lane & 15;     // row within half (M for A, N for B/C)
    const int hi   = lane >> 4;     // lane half select
    const int n0   = (threadIdx.x >> 5) * 16;
    const long row0 = (long)blockIdx.x * 16;
    const int n = n0 + lr;
    const bool nok = (n < Fo);
    // branchless tail handling: clamp the address, mask the value
    const float bmask = nok ? 1.0f : 0.0f;
    const float* Wp = W + (nok ? n : (Fo - 1));

    // low 32 bits of the generic address of a __shared__ object are the LDS
    // byte offset (flat LDS-aperture mapping) -> exactly what D#.lds_addr wants
    const unsigned lds_base = (unsigned)(uintptr_t)&lds[0];

    v8f acc = {};

    for (int kc = 0; kc < K; kc += 64) {
        if (threadIdx.x < 32) {
            // Tensor DMA descriptor: 2D tile, 64 elems x 16 rows, 4B elements
            unsigned long long ga =
                (unsigned long long)(uintptr_t)(A + (size_t)row0 * lda + kc);
            u32x4 g0;
            i32x8 g1;
            i32x4 gz4 = {0, 0, 0, 0};
            i32x8 gz8 = {0, 0, 0, 0, 0, 0, 0, 0};
            g0[0] = 1u;                                   // count=1 (valid D#)
            g0[1] = lds_base;                             // lds_addr (bytes)
            g0[2] = (unsigned)ga;                         // global_addr[31:0]
            g0[3] = (unsigned)((ga >> 32) & 0x1FFFFFFull) // global_addr[56:32]
                  | (2u << 30);                           // type=2 ("image")
            g1[0] = (2 << 16)      // data_size = 2 -> 4 bytes
                  | (1 << 20)      // pad_enable
                  | (5 << 22)      // pad_interval code 5 -> every 64 DWORDs
                  | (3 << 25);     // pad_amount  code 3 -> 4 DWORDs
            g1[1] = (64 << 16);    // tensor_dim0 = 64  (bits 79:48)
            g1[2] = (16 << 16);    // tensor_dim1 = 16  (bits 111:80)
            g1[3] = (64 << 16);    // tile_dim0   = 64  (bits 127:112)
            g1[4] = 16;            // tile_dim1   = 16  (bits 143:128)
            g1[5] = lda;           // tensor_dim0_stride[31:0] (elements)
            g1[6] = 0;             // stride0[47:32]=0, tensor_dim1_stride lo=0
            g1[7] = 0;
            __builtin_amdgcn_tensor_load_to_lds(g0, g1, gz4, gz4, gz8, 0);
            __builtin_amdgcn_s_wait_tensorcnt(0);
        }
        __syncthreads();

#pragma unroll
        for (int k = 0; k < 64; k += 4) {
            // A 16x4 frag: lanes 0-15 -> K = k,k+1 ; lanes 16-31 -> K = k+2,k+3
            v2f a, b;
            a.x = lds[lr * 68 + k + hi * 2 + 0];
            a.y = lds[lr * 68 + k + hi * 2 + 1];
            // B 4x16 frag: VGPR0 holds K rows {k | k+2}, VGPR1 holds {k+1 | k+3}
            int kb = kc + k + hi * 2;
            b.x = Wp[(size_t)(kb + 0) * Fo] * bmask;
            b.y = Wp[(size_t)(kb + 1) * Fo] * bmask;
            acc = __builtin_amdgcn_wmma_f32_16x16x4_f32(
                false, a, false, b, (short)0, acc, false, false);
        }
        __syncthreads();
    }

    if (nok) {
        float bv = bias[n];
#pragma unroll
        for (int v = 0; v < 8; ++v) {
            // C/D layout: VGPR v -> M = v (lanes 0-15) or v+8 (lanes 16-31)
            H[(size_t)(row0 + v + hi * 8) * Fo + n] = acc[v] + bv;
        }
    }
}

// ---------------------------------------------------------------- launcher

extern "C" void kernel_launch(void* const* d_in, const int* in_sizes, int n_in,
                              void* d_out, int out_size, void* d_ws, size_t ws_size,
                              hipStream_t stream) {
    const float* feat = (const float*)d_in[0];
    const int*   src  = (const int*)d_in[1];
    const int*   dst  = (const int*)d_in[2];
    const float* W1   = (const float*)d_in[3];
    const float* b1   = (const float*)d_in[4];
    const float* W2   = (const float*)d_in[5];
    const float* b2   = (const float*)d_in[6];
    const float* W3   = (const float*)d_in[7];
    const float* b3   = (const float*)d_in[8];
    float* out = (float*)d_out;

    const int N = NNODES;
    const int E = in_sizes[1];

    float* ws   = (float*)d_ws;
    float* nrm  = ws;                          // N floats
    float* Xc   = ws + 50048;                  // N * 384 floats (concat buffer)
    float* Ha   = Xc + (size_t)N * 384;        // N * 64
    float* Hb   = Ha + (size_t)N * 64;         // N * 64

    // ---- symmetric norm from in-degrees
    cheb_fill_zero<<<256, 256, 0, stream>>>(nrm, (long)N);
    cheb_degree<<<(E + 255) / 256, 256, 0, stream>>>(dst, nrm, E);
    cheb_norm<<<(N + 255) / 256, 256, 0, stream>>>(nrm, N);

    // ---- one ChebConv layer
    auto layer = [&](const float* X, int F, const float* W, const float* b,
                     int Fo, float* H) {
        const int ld = 3 * F;
        const int lg = (F == 128) ? 5 : 4;     // log2(F/4)
        const long nodeTot = (long)N * (F >> 2);
        const long edgeTot = (long)E * (F >> 2);

        cheb_fill_zero<<<2048, 256, 0, stream>>>(Xc, (long)N * ld);
        // T0 = X
        cheb_copy_block<<<(int)((nodeTot + 255) / 256), 256, 0, stream>>>(
            X, F, Xc, ld, nodeTot, lg);
        // agg1 = segsum(X[src]*norm[src]) into T1 block
        cheb_scatter<<<(int)((edgeTot + 255) / 256), 256, 0, stream>>>(
            X, F, src, dst, nrm, Xc + F, ld, edgeTot, lg);
        // T1 = -norm * agg1
        cheb_finalize_t1<<<(int)((nodeTot + 255) / 256), 256, 0, stream>>>(
            Xc + F, ld, nrm, nodeTot, lg);
        // agg2 = segsum(T1[src]*norm[src]) into T2 block
        cheb_scatter<<<(int)((edgeTot + 255) / 256), 256, 0, stream>>>(
            Xc + F, ld, src, dst, nrm, Xc + 2 * F, ld, edgeTot, lg);
        // T2 = -2*norm*agg2 - T0
        cheb_finalize_t2<<<(int)((nodeTot + 255) / 256), 256, 0, stream>>>(
            Xc + 2 * F, Xc, ld, nrm, nodeTot, lg);
        // H = [T0|T1|T2] @ W + b
        const int nwaves = (Fo + 15) / 16;
        cheb_gemm_wmma<<<N / 16, 32 * nwaves, 0, stream>>>(
            Xc, ld, 3 * F, W, b, H, Fo);
    };

    layer(feat, 128, W1, b1, 64, Ha);
    layer(Ha,    64, W2, b2, 64, Hb);
    layer(Hb,    64, W3, b3, 40, out);
}